// KalmanFilter_34626026341058
// MI455X (gfx1250) — compile-verified
//
#include <hip/hip_runtime.h>

typedef float v2f __attribute__((ext_vector_type(2)));
typedef float v8f __attribute__((ext_vector_type(8)));

namespace {

constexpr int SD = 128;          // state dim
constexpr int OD = 64;           // obs dim
constexpr int T_STEPS = 2048;
constexpr int NTHR = 1024;       // 32 wave32s -> fills one WGP
constexpr int NW = NTHR / 32;
constexpr float JIT = 1e-6f;
constexpr float LOG2PI = 1.8378770664093453f;

// ---- dynamic LDS layout (float offsets) ----
constexpr int OFF_A   = 0;       // A         128x128
constexpr int OFF_P   = 16384;   // P         128x128 (persistent state)
constexpr int OFF_T1  = 32768;   // A@P temp; later Saug(64x128)@+0, K(128x64)@+8192
constexpr int OFF_T2  = 49152;   // P_pred    128x128
constexpr int OFF_CP  = 65536;   // C@P_pred  64x128
constexpr int OFF_Z    = 73728;  // z      128
constexpr int OFF_ZP   = 73856;  // z_pred 128
constexpr int OFF_INNO = 73984;  // innov  64
constexpr int OFF_WV   = 74048;  // Sinv@innov 64
constexpr int OFF_FC   = 74112;  // GJ column factors 64
constexpr int OFF_Y    = 74176;  // y_t (nan-safed) 64
constexpr int OFF_ROWP = 74240;  // GJ scaled pivot row 128
constexpr int LDS_FLOATS = 74368; // ~297.5 KB of the 320 KB WGP LDS

// ---- WMMA f32 16x16x4 fragment helpers (wave32 layouts per ISA 7.12.2) ----
// A-frag 16x4: lanes 0-15 hold K={kb,kb+1}, lanes 16-31 hold K={kb+2,kb+3}
__device__ __forceinline__ v2f frag_a(const float* p, int ld, int trans,
                                      int mr, int kb, int lane) {
  int m = mr + (lane & 15);
  int k = kb + ((lane >> 4) << 1);
  v2f a;
  if (!trans) { a.x = p[m * ld + k];     a.y = p[m * ld + k + 1]; }
  else        { a.x = p[k * ld + m];     a.y = p[(k + 1) * ld + m]; }
  return a;
}
// B-frag 4x16: lanes 0-15 N=lane with K={kb,kb+1}; lanes 16-31 K={kb+2,kb+3}
__device__ __forceinline__ v2f frag_b(const float* p, int ld, int trans,
                                      int kb, int nc, int lane) {
  int n = nc + (lane & 15);
  int k = kb + ((lane >> 4) << 1);
  v2f b;
  if (!trans) { b.x = p[k * ld + n];     b.y = p[(k + 1) * ld + n]; }
  else        { b.x = p[n * ld + k];     b.y = p[n * ld + k + 1]; }
  return b;
}

// Tiled D = A*B (+Cinit) via V_WMMA_F32_16X16X4_F32.
// Software-pipelined k-loop: fragment k+1 is issued before wmma k so the
// DS loads stay in flight under the matrix op (partial s_wait_dscnt).
__device__ __forceinline__ void gemm_tiles(const float* Ab, int lda, int ta,
                                           const float* Bb, int ldb, int tb,
                                           const float* Cb, int ldc,   // nullptr -> 0
                                           float* Db, int ldd,
                                           int M, int N, int K, int tid) {
  const int lane = tid & 31;
  const int wave = tid >> 5;
  const int tn = N >> 4;
  const int ntiles = (M >> 4) * tn;
  for (int tt = wave; tt < ntiles; tt += NW) {
    const int mr = (tt / tn) << 4;
    const int nc = (tt % tn) << 4;
    const int n  = nc + (lane & 15);
    const int mo = mr + ((lane >> 4) << 3);   // C/D rows: VGPR r -> M=r (+8 hi half)
    v8f acc;
    if (Cb) {
#pragma unroll
      for (int r = 0; r < 8; ++r) acc[r] = Cb[(mo + r) * ldc + n];
    } else {
#pragma unroll
      for (int r = 0; r < 8; ++r) acc[r] = 0.0f;
    }
    v2f a = frag_a(Ab, lda, ta, mr, 0, lane);
    v2f b = frag_b(Bb, ldb, tb, 0, nc, lane);
    for (int kb = 4; kb < K; kb += 4) {
      v2f an = frag_a(Ab, lda, ta, mr, kb, lane);   // prefetch next fragments
      v2f bn = frag_b(Bb, ldb, tb, kb, nc, lane);
      acc = __builtin_amdgcn_wmma_f32_16x16x4_f32(false, a, false, b,
                                                  (short)0, acc, false, false);
      a = an; b = bn;
    }
    acc = __builtin_amdgcn_wmma_f32_16x16x4_f32(false, a, false, b,
                                                (short)0, acc, false, false);
#pragma unroll
    for (int r = 0; r < 8; ++r) Db[(mo + r) * ldd + n] = acc[r];
  }
}

__global__ __launch_bounds__(NTHR)
void kalman_kernel(const float* __restrict__ y,  const float* __restrict__ Ag,
                   const float* __restrict__ Cg, const float* __restrict__ Qg,
                   const float* __restrict__ Rg, const float* __restrict__ z0,
                   const float* __restrict__ P0, float* __restrict__ out) {
  extern __shared__ float sm[];
  __shared__ float s_ll, s_logdet;
  __shared__ int s_flag;
  const int tid = threadIdx.x;
  const int lane = tid & 31;
  const int wave = tid >> 5;

  float* Abuf = sm + OFF_A;
  float* Pbuf = sm + OFF_P;
  float* T1   = sm + OFF_T1;
  float* Saug = T1;              // 64 x 128 augmented [S | I]
  float* Kbuf = T1 + 8192;       // 128 x 64 Kalman gain
  float* T2   = sm + OFF_T2;     // P_pred
  float* CPb  = sm + OFF_CP;     // 64 x 128
  float* zv   = sm + OFF_Z;
  float* zp   = sm + OFF_ZP;
  float* inno = sm + OFF_INNO;
  float* wv   = sm + OFF_WV;
  float* fc   = sm + OFF_FC;
  float* ysh  = sm + OFF_Y;
  float* rowp = sm + OFF_ROWP;

  float* out_states = out;
  float* out_cov    = out + (size_t)T_STEPS * SD;
  float* out_ll     = out + (size_t)T_STEPS * SD + (size_t)T_STEPS * SD * SD;

  for (int i = tid; i < SD * SD; i += NTHR) Abuf[i] = Ag[i];
  if (tid == 0) s_ll = 0.0f;
  __syncthreads();

  for (int t = 0; t < T_STEPS; ++t) {
    // ---- observation load + NaN mask ----
    if (tid == 0) s_flag = 0;
    __syncthreads();
    if (tid < OD) {
      float yvl = y[t * OD + tid];
      if (yvl != yvl) { s_flag = 1; yvl = 0.0f; }
      ysh[tid] = yvl;
    }
    // ---- predict: z_pred, P_pred(T2) ----
    if (t == 0) {
      if (tid < SD) zp[tid] = z0[tid];
      for (int i = tid; i < SD * SD; i += NTHR) T2[i] = P0[i];
      __syncthreads();
    } else {
      if (tid < SD) {
        float acc = 0.0f;
        for (int k = 0; k < SD; ++k) acc += Abuf[tid * SD + k] * zv[k];
        zp[tid] = acc;
      }
      __syncthreads();
      gemm_tiles(Abuf, SD, 0, Pbuf, SD, 0, nullptr, 0, T1, SD, SD, SD, SD, tid); // T1=A@P
      __syncthreads();
      gemm_tiles(T1, SD, 0, Abuf, SD, 1, Qg, SD, T2, SD, SD, SD, SD, tid); // T2=T1@A^T+Q
      __syncthreads();
    }
    // ---- CP = C @ P_pred  (PCt = CP^T by symmetry of P_pred) ----
    gemm_tiles(Cg, SD, 0, T2, SD, 0, nullptr, 0, CPb, SD, OD, SD, SD, tid);
    __syncthreads();
    // ---- Saug left = CP @ C^T + R ; right = I ; diag += jitter ----
    gemm_tiles(CPb, SD, 0, Cg, SD, 1, Rg, OD, Saug, 2 * OD, OD, OD, SD, tid);
    __syncthreads();
    for (int e = tid; e < OD * OD; e += NTHR) {
      int r = e >> 6, c = e & 63;
      Saug[r * 128 + 64 + c] = (r == c) ? 1.0f : 0.0f;
    }
    if (tid < OD) Saug[tid * 128 + tid] += JIT;
    if (tid == 0) s_logdet = 0.0f;
    __syncthreads();
    // ---- Gauss-Jordan [S|I] -> [I|Sinv], logdet from pivots; 2 barriers/pivot ----
    for (int p = 0; p < OD; ++p) {
      // pass 1: stage scaled pivot row + column factors (reads only from Saug)
      if (tid < 128) {
        float pv = Saug[p * 128 + p];          // each thread reads pivot itself
        rowp[tid] = Saug[p * 128 + tid] * (1.0f / pv);
      }
      if (tid >= 128 && tid < 128 + OD) {
        int r = tid - 128;
        fc[r] = (r == p) ? 0.0f : Saug[r * 128 + p];
      }
      if (tid == 0) s_logdet += __logf(fabsf(Saug[p * 128 + p]));
      __syncthreads();
      // pass 2: rank-1 update + write scaled pivot row
      for (int e = tid; e < OD * 128; e += NTHR) {
        int r = e >> 7, c = e & 127;
        float rp = rowp[c];
        Saug[r * 128 + c] = (r == p) ? rp : (Saug[r * 128 + c] - fc[r] * rp);
      }
      __syncthreads();
    }
    // ---- innovation, quad form, log-likelihood ----
    if (tid < OD) {
      float acc = 0.0f;
      for (int k = 0; k < SD; ++k) acc += Cg[tid * SD + k] * zp[k];
      inno[tid] = ysh[tid] - acc;
    }
    __syncthreads();
    if (tid < OD) {
      const float* Sinv = Saug + 64;
      float acc = 0.0f;
      for (int j = 0; j < OD; ++j) acc += Sinv[tid * 128 + j] * inno[j];
      wv[tid] = acc;
    }
    __syncthreads();
    if (tid == 0) {
      float q = 0.0f;
      for (int j = 0; j < OD; ++j) q += inno[j] * wv[j];
      if (!s_flag) s_ll += -0.5f * (s_logdet + q + (float)OD * LOG2PI);
    }
    // ---- K = CP^T @ Sinv  (128x64x64) ----
    gemm_tiles(CPb, SD, 1, Saug + 64, 128, 0, nullptr, 0, Kbuf, OD, SD, OD, OD, tid);
    __syncthreads();
    // ---- z update + state output ----
    if (tid < SD) {
      float acc = 0.0f;
      for (int j = 0; j < OD; ++j) acc += Kbuf[tid * OD + j] * inno[j];
      float zn = s_flag ? zp[tid] : (zp[tid] + acc);
      zv[tid] = zn;
      out_states[(size_t)t * SD + tid] = zn;
    }
    // ---- covariance update: P = P_pred - K@CP - jit*K@K^T (== Joseph form) ----
    float* covout = out_cov + (size_t)t * SD * SD;
    if (!s_flag) {
      for (int tt = wave; tt < 64; tt += NW) {           // 8x8 tiles of 128x128
        const int mr = (tt >> 3) << 4;
        const int nc = (tt & 7) << 4;
        const int n  = nc + (lane & 15);
        const int mo = mr + ((lane >> 4) << 3);
        v8f acc;
#pragma unroll
        for (int r = 0; r < 8; ++r) acc[r] = 0.0f;
        v2f a  = frag_a(Kbuf, OD, 0, mr, 0, lane);
        v2f b  = frag_b(CPb, SD, 0, 0, nc, lane);
        v2f b2 = frag_b(Kbuf, OD, 1, 0, nc, lane);
        for (int kb = 4; kb < OD; kb += 4) {
          v2f an  = frag_a(Kbuf, OD, 0, mr, kb, lane);   // prefetch next fragments
          v2f bn  = frag_b(CPb, SD, 0, kb, nc, lane);
          v2f b2n = frag_b(Kbuf, OD, 1, kb, nc, lane);
          acc = __builtin_amdgcn_wmma_f32_16x16x4_f32(false, a, false, b,
                                                      (short)0, acc, false, false);
          v2f bj = b2; bj.x *= JIT; bj.y *= JIT;         // jit * K^T
          acc = __builtin_amdgcn_wmma_f32_16x16x4_f32(false, a, false, bj,
                                                      (short)0, acc, false, false);
          a = an; b = bn; b2 = b2n;
        }
        acc = __builtin_amdgcn_wmma_f32_16x16x4_f32(false, a, false, b,
                                                    (short)0, acc, false, false);
        v2f bj = b2; bj.x *= JIT; bj.y *= JIT;
        acc = __builtin_amdgcn_wmma_f32_16x16x4_f32(false, a, false, bj,
                                                    (short)0, acc, false, false);
#pragma unroll
        for (int r = 0; r < 8; ++r) {
          int idx = (mo + r) * SD + n;
          float v = T2[idx] - acc[r];
          Pbuf[idx] = v;
          covout[idx] = v;
        }
      }
    } else {
      for (int i = tid; i < SD * SD; i += NTHR) {
        float v = T2[i];
        Pbuf[i] = v;
        covout[i] = v;
      }
    }
    __syncthreads();
  }
  if (tid == 0) *out_ll = s_ll;
}

} // namespace

extern "C" void kernel_launch(void* const* d_in, const int* in_sizes, int n_in,
                              void* d_out, int out_size, void* d_ws, size_t ws_size,
                              hipStream_t stream) {
  (void)in_sizes; (void)n_in; (void)out_size; (void)d_ws; (void)ws_size;
  const float* y  = (const float*)d_in[0];
  const float* A  = (const float*)d_in[1];
  const float* C  = (const float*)d_in[2];
  const float* Q  = (const float*)d_in[3];
  const float* R  = (const float*)d_in[4];
  const float* z0 = (const float*)d_in[5];
  const float* P0 = (const float*)d_in[6];
  float* out = (float*)d_out;

  const size_t shmem = (size_t)LDS_FLOATS * sizeof(float); // ~297.5 KB of 320 KB WGP LDS
  hipFuncSetAttribute((const void*)kalman_kernel,
                      hipFuncAttributeMaxDynamicSharedMemorySize, (int)shmem);
  kalman_kernel<<<1, NTHR, shmem, stream>>>(y, A, C, Q, R, z0, P0, out);
}